// IoUWeightedSmoothL1Loss_67808943669848
// MI455X (gfx1250) — compile-verified
//
#include <hip/hip_runtime.h>
#include <math.h>

typedef __attribute__((ext_vector_type(2))) float v2f;
typedef __attribute__((ext_vector_type(8))) float v8f;

#define EPSF    1.1920928955078125e-7f   // np.finfo(float32).eps
#define NPTS    18
#define NPAIR   630                       // C(36,2)
#define NTILE   40                        // ceil(630/16)
#define NB      16
#define NK      64
#define NROWS   (NB*NK)                   // 1024
#define HW      65536                     // 256*256

// ---------------------------------------------------------------------------
// Min-area rectangle of the symmetric 36-point set {(px,py)_i, -(px,py)_i}.
// Projections of all candidate edge directions evaluated as a
// (points 16 x K2) x (K2 x candidates 16) GEMM with V_WMMA_F32_16X16X4_F32.
// Whole wave (32 lanes) cooperates; corners written to LDS by lane 0.
// ---------------------------------------------------------------------------
__device__ __forceinline__ void min_area_rect(const float* px, const float* py,
                                              const unsigned short* pair,
                                              float (*corners)[2]) {
  const int lane = threadIdx.x;
  const int nlo  = lane & 15;

  // A (16x4 f32 layout): VGPR0 = K0 (lanes 0-15) | K2 (lanes 16-31, zero pad)
  //                      VGPR1 = K1 (lanes 0-15) | K3 (lanes 16-31, zero pad)
  v2f A  = {0.f, 0.f};
  v2f Ar = {0.f, 0.f};                       // rotated points -> normal axis
  if (lane < 16) {
    A[0]  = px[lane];  A[1]  = py[lane];
    Ar[0] = py[lane];  Ar[1] = -px[lane];    // v.p == u.(py,-px)
  }
  const float p16x = px[16], p16y = py[16], p17x = px[17], p17y = py[17];

  float bestArea = __builtin_inff();
  int   bestT    = 0x7fffffff;
  float bUx = 0.f, bUy = 0.f, bUm = 0.f, bVm = 0.f;

  for (int tile = 0; tile < NTILE; ++tile) {
    const int t = tile * 16 + nlo;           // global pair index (triu order)
    float ux = 0.f, uy = 0.f;
    bool  valid = false;
    if (t < NPAIR) {
      const int pr = pair[t];
      const int ii = pr & 0xff, jj = pr >> 8;
      const float ax = (ii < NPTS) ? px[ii] : -px[ii - NPTS];
      const float ay = (ii < NPTS) ? py[ii] : -py[ii - NPTS];
      const float bx = (jj < NPTS) ? px[jj] : -px[jj - NPTS];
      const float by = (jj < NPTS) ? py[jj] : -py[jj - NPTS];
      const float dx = bx - ax, dy = by - ay;
      const float nrm = sqrtf(dx * dx + dy * dy);
      valid = (nrm > 1e-9f);
      const float inv = 1.0f / fmaxf(nrm, 1e-9f);
      ux = dx * inv; uy = dy * inv;
    }
    // B (4x16 f32): K0/K1 rows in lanes 0-15, K2/K3 rows zero (lanes 16-31)
    v2f Bm = {0.f, 0.f};
    if (lane < 16) { Bm[0] = ux; Bm[1] = uy; }

    v8f cz = {};
    // D[m,n] = p_m . u_n   and   D[m,n] = v_n . p_m
    v8f Cu = __builtin_amdgcn_wmma_f32_16x16x4_f32(false, A,  false, Bm,
                                                   (short)0, cz, false, false);
    v8f Cv = __builtin_amdgcn_wmma_f32_16x16x4_f32(false, Ar, false, Bm,
                                                   (short)0, cz, false, false);
    // symmetric point set => umin = -umax, umax = max |projection|
    float um = 0.f, vm = 0.f;
#pragma unroll
    for (int r = 0; r < 8; ++r) {
      um = fmaxf(um, fabsf(Cu[r]));
      vm = fmaxf(vm, fabsf(Cv[r]));
    }
    // lane n: points 0-7 ; lane n+16: points 8-15 of the same candidate n
    um = fmaxf(um, __shfl_xor(um, 16, 32));
    vm = fmaxf(vm, __shfl_xor(vm, 16, 32));
    // points 16,17 (M overflow of the 16x16 tile), 2 scalar FMAs each
    um = fmaxf(um, fabsf(ux * p16x + uy * p16y));
    um = fmaxf(um, fabsf(ux * p17x + uy * p17y));
    vm = fmaxf(vm, fabsf(ux * p16y - uy * p16x));
    vm = fmaxf(vm, fabsf(ux * p17y - uy * p17x));

    const float area = valid ? 4.f * um * vm : __builtin_inff();
    if (area < bestArea || (area == bestArea && t < bestT)) {
      bestArea = area; bestT = t; bUx = ux; bUy = uy; bUm = um; bVm = vm;
    }
  }

  // wave argmin (tie -> lowest pair index, matching jnp.argmin)
  for (int off = 16; off; off >>= 1) {
    const float oA  = __shfl_xor(bestArea, off, 32);
    const int   oT  = __shfl_xor(bestT,    off, 32);
    const float oUx = __shfl_xor(bUx,      off, 32);
    const float oUy = __shfl_xor(bUy,      off, 32);
    const float oUm = __shfl_xor(bUm,      off, 32);
    const float oVm = __shfl_xor(bVm,      off, 32);
    if (oA < bestArea || (oA == bestArea && oT < bestT)) {
      bestArea = oA; bestT = oT; bUx = oUx; bUy = oUy; bUm = oUm; bVm = oVm;
    }
  }

  if (lane == 0) {
    const float vx = -bUy, vy = bUx;                 // v = rot90(u)
    const float cu[4] = {-bUm,  bUm,  bUm, -bUm};
    const float cv[4] = {-bVm, -bVm,  bVm,  bVm};
    float cx[4], cy[4], an[4];
#pragma unroll
    for (int i = 0; i < 4; ++i) {
      cx[i] = cu[i] * bUx + cv[i] * vx;
      cy[i] = cu[i] * bUy + cv[i] * vy;
      an[i] = atan2f(cy[i], cx[i]);
    }
    int idx[4] = {0, 1, 2, 3};                       // stable argsort by angle
    for (int i = 1; i < 4; ++i) {
      int key = idx[i]; float ka = an[key]; int j = i - 1;
      while (j >= 0 && an[idx[j]] > ka) { idx[j + 1] = idx[j]; --j; }
      idx[j + 1] = key;
    }
#pragma unroll
    for (int i = 0; i < 4; ++i) { corners[i][0] = cx[idx[i]]; corners[i][1] = cy[idx[i]]; }
  }
}

__device__ __forceinline__ float shoelace_n(float (*p)[2], int n) {
  float s = 0.f;
  for (int i = 0; i < n; ++i) {
    const int j = (i + 1 == n) ? 0 : i + 1;
    s += p[i][0] * p[j][1] - p[i][1] * p[j][0];
  }
  return 0.5f * fabsf(s);
}

// Sutherland-Hodgman clip (keep side >= 0, matching reference), then IoU.
__device__ float poly_iou(float (*ca)[2], float (*cb)[2],
                          float (*poly)[2], float (*outp)[2]) {
  int n = 4;
  for (int i = 0; i < 4; ++i) { poly[i][0] = ca[i][0]; poly[i][1] = ca[i][1]; }
  for (int e = 0; e < 4 && n > 0; ++e) {
    const float e1x = cb[e][0],        e1y = cb[e][1];
    const float e2x = cb[(e + 1) & 3][0], e2y = cb[(e + 1) & 3][1];
    const float dex = e2x - e1x, dey = e2y - e1y;
    int m = 0;
    for (int i = 0; i < n; ++i) {
      const int j = (i + 1 == n) ? 0 : i + 1;
      const float sx = poly[i][0], sy = poly[i][1];
      const float ex = poly[j][0], ey = poly[j][1];
      const float ss = dex * (sy - e1y) - dey * (sx - e1x);
      const float se = dex * (ey - e1y) - dey * (ex - e1x);
      const bool sin_ = (ss >= 0.f), ein = (se >= 0.f);
      if (sin_ != ein) {
        float den = ss - se;
        den = (fabsf(den) < 1e-12f) ? 1e-12f : den;
        const float tp = ss / den;
        outp[m][0] = sx + tp * (ex - sx);
        outp[m][1] = sy + tp * (ey - sy);
        ++m;
      }
      if (ein) { outp[m][0] = ex; outp[m][1] = ey; ++m; }
    }
    n = m;
    for (int i = 0; i < n; ++i) { poly[i][0] = outp[i][0]; poly[i][1] = outp[i][1]; }
  }
  const float inter = (n >= 3) ? shoelace_n(poly, n) : 0.f;
  const float uni = shoelace_n(ca, 4) + shoelace_n(cb, 4) - inter;
  return inter / fmaxf(uni, 1e-12f);
}

// One wave per (b,k) row: gather, smooth-L1, two min-area rects (WMMA),
// polygon IoU, per-row partials into workspace (overwrite, no atomics).
__global__ __launch_bounds__(32)
void IoUWeightedSmoothL1_main(const float* __restrict__ output,
                              const int*   __restrict__ mask,
                              const int*   __restrict__ ind,
                              const float* __restrict__ target,
                              float* __restrict__ ws_reg,
                              float* __restrict__ ws_iou) {
  __shared__ float s_pred[NPTS], s_tgt[NPTS];
  __shared__ float s_px[NPTS], s_py[NPTS];
  __shared__ float s_ca[4][2], s_cb[4][2];
  __shared__ float s_poly[16][2], s_out[16][2];
  __shared__ unsigned short s_pair[NPAIR];

  const int lane = threadIdx.x;
  const int row  = blockIdx.x;          // 0..1023
  const int b    = row >> 6;            // K = 64

  // triu_indices(36, k=1) pair table, packed (ii | jj<<8)
  for (int t = lane; t < NPAIR; t += 32) {
    int tt = t, ii = 0;
    while (tt >= 35 - ii) { tt -= 35 - ii; ++ii; }
    s_pair[t] = (unsigned short)(ii | ((ii + 1 + tt) << 8));
  }
  // gather pred (output[b, n, pix]) and target row
  if (lane < NPTS) {
    const int pix = ind[row];
    s_pred[lane] = output[(size_t)(b * NPTS + lane) * HW + pix];
    s_tgt[lane]  = target[(size_t)row * NPTS + lane];
  }
  __syncthreads();

  // smooth-L1 (mean over 18 dims), wave butterfly sum
  float v = 0.f;
  if (lane < NPTS) {
    const float d  = s_pred[lane] - s_tgt[lane];
    const float ad = fabsf(d);
    v = (ad < 1.f) ? 0.5f * d * d : ad - 0.5f;
  }
  for (int off = 16; off; off >>= 1) v += __shfl_xor(v, off, 32);
  const float sl1 = v * (1.0f / (float)NPTS);

  // rect of pred
  if (lane < NPTS) {
    const float ang = (float)lane * 0.17453292519943295f;   // pi/18
    s_px[lane] = s_pred[lane] * cosf(ang);
    s_py[lane] = s_pred[lane] * sinf(ang);
  }
  __syncthreads();
  min_area_rect(s_px, s_py, s_pair, s_ca);
  __syncthreads();

  // rect of target
  if (lane < NPTS) {
    const float ang = (float)lane * 0.17453292519943295f;
    s_px[lane] = s_tgt[lane] * cosf(ang);
    s_py[lane] = s_tgt[lane] * sinf(ang);
  }
  __syncthreads();
  min_area_rect(s_px, s_py, s_pair, s_cb);
  __syncthreads();

  if (lane == 0) {
    const float iou   = poly_iou(s_ca, s_cb, s_poly, s_out);
    const float mf    = (float)mask[row];
    const float alpha = -logf(fabsf(iou) + EPSF);
    const float li    = alpha * sl1 / (fabsf(sl1) + EPSF);
    ws_reg[row] = sl1 * mf;
    ws_iou[row] = li * mf;
  }
}

// Deterministic fixed-order reduction (graph-replay safe, no atomics).
__global__ __launch_bounds__(256)
void IoUWeightedSmoothL1_reduce(const float* __restrict__ ws_reg,
                                const float* __restrict__ ws_iou,
                                const int*   __restrict__ mask,
                                float* __restrict__ out) {
  __shared__ float sr[256], si[256];
  __shared__ int   sc[256];
  const int tid = threadIdx.x;
  float r = 0.f, o = 0.f; int c = 0;
  for (int i = tid; i < NROWS; i += 256) { r += ws_reg[i]; o += ws_iou[i]; c += mask[i]; }
  sr[tid] = r; si[tid] = o; sc[tid] = c;
  __syncthreads();
  for (int s = 128; s; s >>= 1) {
    if (tid < s) { sr[tid] += sr[tid + s]; si[tid] += si[tid + s]; sc[tid] += sc[tid + s]; }
    __syncthreads();
  }
  if (tid == 0) {
    const float cnt = fmaxf((float)sc[0], 1.0f);
    out[0] = sr[0] / cnt;   // loss_reg
    out[1] = si[0] / cnt;   // loss_iou
  }
}

extern "C" void kernel_launch(void* const* d_in, const int* in_sizes, int n_in,
                              void* d_out, int out_size, void* d_ws, size_t ws_size,
                              hipStream_t stream) {
  (void)in_sizes; (void)n_in; (void)out_size; (void)ws_size;
  const float* output = (const float*)d_in[0];
  const int*   mask   = (const int*)d_in[1];
  const int*   ind    = (const int*)d_in[2];
  const float* target = (const float*)d_in[3];
  float* ws_reg = (float*)d_ws;
  float* ws_iou = ws_reg + NROWS;

  IoUWeightedSmoothL1_main<<<NROWS, 32, 0, stream>>>(output, mask, ind, target,
                                                     ws_reg, ws_iou);
  IoUWeightedSmoothL1_reduce<<<1, 256, 0, stream>>>(ws_reg, ws_iou, mask,
                                                    (float*)d_out);
}